// GL_SSGConv_3l_128h_w_a09_k1_44753559224356
// MI455X (gfx1250) — compile-verified
//
#include <hip/hip_runtime.h>
#include <math.h>

typedef float v2f __attribute__((ext_vector_type(2)));
typedef float v8f __attribute__((ext_vector_type(8)));

#define N_NODES  50000
#define N_EDGES  600000
#define N_FEAT   128
#define HIDDEN   128
#define N_CLASSES 10
#define ALPHA    0.9f
#define M_TILES  (N_NODES / 16)   // 3125, exact

// ---------------- graph-norm kernels ----------------

__global__ void k_init_deg(float* __restrict__ deg) {
    int i = blockIdx.x * blockDim.x + threadIdx.x;
    if (i < N_NODES) deg[i] = 1.0f;   // self-loop weight
}

__global__ void k_accum_deg(const int* __restrict__ dst, const float* __restrict__ w,
                            float* __restrict__ deg) {
    int e = blockIdx.x * blockDim.x + threadIdx.x;
    if (e < N_EDGES) atomicAdd(&deg[dst[e]], w[e]);
}

__global__ void k_dinv(const float* __restrict__ deg, float* __restrict__ dinv) {
    int i = blockIdx.x * blockDim.x + threadIdx.x;
    if (i < N_NODES) {
        float d = deg[i];
        dinv[i] = d > 0.0f ? rsqrtf(d) : 0.0f;
    }
}

// norm_e = (1-alpha) * dinv[src] * w * dinv[dst]   ((1-alpha) folded in)
__global__ void k_norm(const int* __restrict__ src, const int* __restrict__ dst,
                       const float* __restrict__ w, const float* __restrict__ dinv,
                       float* __restrict__ norm) {
    int e = blockIdx.x * blockDim.x + threadIdx.x;
    if (e < N_EDGES) norm[e] = (1.0f - ALPHA) * dinv[src[e]] * w[e] * dinv[dst[e]];
}

// h = (alpha + (1-alpha)*dinv^2) * x   (self loop + identity blend)
__global__ void k_init_h(const float* __restrict__ x, const float* __restrict__ dinv,
                         float* __restrict__ h) {
    int i = blockIdx.x * blockDim.x + threadIdx.x;
    if (i < N_NODES * N_FEAT) {
        int node = i >> 7;   // /128
        float di = dinv[node];
        float coef = ALPHA + (1.0f - ALPHA) * di * di;
        h[i] = coef * x[i];
    }
}

// one wave (32 lanes) per edge; each lane handles 4 features (float4 gather + 4 atomics)
__global__ void k_scatter(const int* __restrict__ src, const int* __restrict__ dst,
                          const float* __restrict__ norm, const float* __restrict__ x,
                          float* __restrict__ h) {
    int e    = (blockIdx.x * blockDim.x + threadIdx.x) >> 5;
    int lane = threadIdx.x & 31;
    if (e >= N_EDGES) return;
    int   s  = src[e];
    int   d  = dst[e];
    float nw = norm[e];
    const float4 v = ((const float4*)(x + (size_t)s * N_FEAT))[lane];
    float* hd = h + (size_t)d * N_FEAT + lane * 4;
    atomicAdd(hd + 0, nw * v.x);
    atomicAdd(hd + 1, nw * v.y);
    atomicAdd(hd + 2, nw * v.z);
    atomicAdd(hd + 3, nw * v.w);
}

// ---------------- WMMA GEMM: out[M,FOUT] = h[M,128] @ W[128,FOUT] + b (+ELU) ---------
//
// W staged TRANSPOSED in LDS: sW[n * SK + k]; SK = 132 (== 4 mod 64) so the per-lane
// 64-bit B-fragment fetches are bank-conflict free (lanes 0-15 -> banks {4n,4n+1},
// lanes 16-31 -> {4n+2,4n+3}).  Per-tile base pointers are hoisted so in-loop LDS
// addresses are base + k*4 bytes (fits the DS immediate); all NT fragment loads are
// issued before the WMMA burst so LDS latency overlaps matrix-core execution.

template <int FOUT, bool DO_ELU>
__global__ void k_gemm(const float* __restrict__ h, const float* __restrict__ W,
                       const float* __restrict__ bias, float* __restrict__ out) {
    constexpr int NT  = (FOUT + 15) / 16;   // N tiles of 16
    constexpr int NR  = NT * 16;            // padded rows in transposed LDS image
    constexpr int SK  = 128 + 4;            // K stride (words), == 4 mod 64

    extern __shared__ float sW[];           // NR x SK floats
    for (int i = threadIdx.x; i < 128 * NR; i += blockDim.x) {
        int k = i / NR;                     // coalesced global read (n fastest)
        int n = i - k * NR;
        sW[n * SK + k] = (NR == FOUT || n < FOUT) ? W[k * FOUT + n] : 0.0f;
    }
    __syncthreads();

    const int waveInBlk = threadIdx.x >> 5;
    const int lane      = threadIdx.x & 31;
    const int tile_m    = blockIdx.x * (blockDim.x >> 5) + waveInBlk;
    if (tile_m >= M_TILES) return;          // no barriers after this point

    const int half = lane >> 4;             // 0: K+0/1,  1: K+2/3
    const int l16  = lane & 15;
    const int m    = tile_m * 16 + l16;     // A row handled by this lane
    const int kh   = half * 2;

    // hoisted per-tile LDS base pointers: in-loop address = bptr[t] + k  (imm offset)
    const float* bptr[NT];
    #pragma unroll
    for (int t = 0; t < NT; ++t) bptr[t] = sW + (t * 16 + l16) * SK + kh;

    v8f acc[NT];
    #pragma unroll
    for (int t = 0; t < NT; ++t) acc[t] = (v8f){};

    const float* __restrict__ hrow = h + (size_t)m * 128;
    for (int k = 0; k < 128; k += 4) {
        v2f a;
        a.x = hrow[k + kh];
        a.y = hrow[k + kh + 1];

        v2f b[NT];                          // batch all fragment loads first
        #pragma unroll
        for (int t = 0; t < NT; ++t) b[t] = *(const v2f*)(bptr[t] + k);

        #pragma unroll
        for (int t = 0; t < NT; ++t) {
            // D = A(16x4,f32) x B(4x16,f32) + C  -> v_wmma_f32_16x16x4_f32
            acc[t] = __builtin_amdgcn_wmma_f32_16x16x4_f32(
                false, a, false, b[t], (short)0, acc[t], false, false);
        }
    }

    #pragma unroll
    for (int t = 0; t < NT; ++t) {
        const int n = t * 16 + l16;
        if (n < FOUT) {
            const float bn = bias[n];
            #pragma unroll
            for (int j = 0; j < 8; ++j) {
                const int row = tile_m * 16 + j + 8 * half;   // C/D VGPR layout
                float v = acc[t][j] + bn;
                if (DO_ELU) v = (v > 0.0f) ? v : expm1f(v);
                out[(size_t)row * FOUT + n] = v;
            }
        }
    }
}

// ---------------- launcher ----------------

extern "C" void kernel_launch(void* const* d_in, const int* in_sizes, int n_in,
                              void* d_out, int out_size, void* d_ws, size_t ws_size,
                              hipStream_t stream) {
    const float* x  = (const float*)d_in[0];
    const int*   ei = (const int*)  d_in[1];   // [2][N_EDGES]
    const float* ew = (const float*)d_in[2];
    const float* W1 = (const float*)d_in[3];
    const float* b1 = (const float*)d_in[4];
    const float* W2 = (const float*)d_in[5];
    const float* b2 = (const float*)d_in[6];
    const float* W3 = (const float*)d_in[7];
    const float* b3 = (const float*)d_in[8];
    float* out = (float*)d_out;

    const int* src = ei;
    const int* dst = ei + N_EDGES;

    float* ws   = (float*)d_ws;
    float* deg  = ws;  ws += N_NODES;
    float* dinv = ws;  ws += N_NODES;
    float* norm = ws;  ws += N_EDGES;
    float* bufA = ws;  ws += (size_t)N_NODES * HIDDEN;
    float* bufB = ws;  ws += (size_t)N_NODES * HIDDEN;
    float* bufH = ws;  ws += (size_t)N_NODES * HIDDEN;

    const int tb = 256;
    const int gN  = (N_NODES + tb - 1) / tb;
    const int gE  = (N_EDGES + tb - 1) / tb;
    const int gNF = (N_NODES * N_FEAT + tb - 1) / tb;
    const int gSc = (N_EDGES * 32 + tb - 1) / tb;          // one wave per edge

    const dim3 gemmBlk(128);                                // 4 waves / block
    const int  gemmGrid = (M_TILES + 3) / 4;
    const size_t shH = (size_t)(128) * (128 + 4) * sizeof(float); // NR=128
    const size_t shC = (size_t)(16)  * (128 + 4) * sizeof(float); // NR=16

    // graph normalization (once, reused by all 3 layers)
    k_init_deg <<<gN, tb, 0, stream>>>(deg);
    k_accum_deg<<<gE, tb, 0, stream>>>(dst, ew, deg);
    k_dinv     <<<gN, tb, 0, stream>>>(deg, dinv);
    k_norm     <<<gE, tb, 0, stream>>>(src, dst, ew, dinv, norm);

    // layer 1: x -> bufA
    k_init_h <<<gNF, tb, 0, stream>>>(x, dinv, bufH);
    k_scatter<<<gSc, tb, 0, stream>>>(src, dst, norm, x, bufH);
    k_gemm<HIDDEN, true><<<gemmGrid, gemmBlk, shH, stream>>>(bufH, W1, b1, bufA);

    // layer 2: bufA -> bufB
    k_init_h <<<gNF, tb, 0, stream>>>(bufA, dinv, bufH);
    k_scatter<<<gSc, tb, 0, stream>>>(src, dst, norm, bufA, bufH);
    k_gemm<HIDDEN, true><<<gemmGrid, gemmBlk, shH, stream>>>(bufH, W2, b2, bufB);

    // layer 3: bufB -> out (no ELU)
    k_init_h <<<gNF, tb, 0, stream>>>(bufB, dinv, bufH);
    k_scatter<<<gSc, tb, 0, stream>>>(src, dst, norm, bufB, bufH);
    k_gemm<N_CLASSES, false><<<gemmGrid, gemmBlk, shC, stream>>>(bufH, W3, b3, out);
}